// AutoCorr_58609123721970
// MI455X (gfx1250) — compile-verified
//
#include <hip/hip_runtime.h>
#include <math.h>

typedef __attribute__((ext_vector_type(2))) float v2f;
typedef __attribute__((ext_vector_type(8))) float v8f;

#define NUM_AC    32
#define NWIN      64
#define WLEN      256
#define WSTRIDE   128
#define ROWLEN    8320           // (NWIN-1)*WSTRIDE + WLEN
#define CSTRIDE_M 20             // padded row stride for C scatter (bank-conflict free)
#define CSTRIDE_C 320            // 16 * CSTRIDE_M

__global__ __launch_bounds__(256, 1)
void autocorr_wmma_kernel(const float* __restrict__ x, float* __restrict__ out)
{
    __shared__ __align__(16) float s_row[ROWLEN];     // one (b,s) row
    __shared__ float s_hann[WLEN];                    // Hann table
    __shared__ float s_v[8][288];                     // per-wave windowed signal, zero-padded
    __shared__ float s_c[8][3 * CSTRIDE_C];           // per-wave C_0,C_1,C_2 scatter area

    const int tid  = threadIdx.x;
    const int lane = tid & 31;
    const int wv   = tid >> 5;
    const int col  = lane & 15;
    const int hi   = lane >> 4;
    const int row  = blockIdx.x;                      // b*128 + s

    // np.hanning(256): 0.5 - 0.5*cos(2*pi*t/255)
    if (tid < WLEN) {
        s_hann[tid] = 0.5f - 0.5f * cosf(6.28318530717958647692f * (float)tid * (1.0f / 255.0f));
    }

    // Stage the whole row into LDS with 128-bit loads (row base is 16B aligned).
    {
        const float4* src = reinterpret_cast<const float4*>(x + (size_t)row * ROWLEN);
        float4*       dst = reinterpret_cast<float4*>(s_row);
        for (int i = tid; i < ROWLEN / 4; i += 256) dst[i] = src[i];
    }
    __syncthreads();

    float* vb = s_v[wv];
    float* cb = s_c[wv];

    for (int w = wv; w < NWIN; w += 8) {
        const float* win = s_row + w * WSTRIDE;

        // ---- mean of the raw window (wave32 butterfly reduction) ----
        float ps = 0.0f;
        #pragma unroll
        for (int i = 0; i < 8; ++i) ps += win[i * 32 + lane];
        #pragma unroll
        for (int d = 16; d >= 1; d >>= 1) ps += __shfl_xor(ps, d, 32);
        const float mean = ps * (1.0f / 256.0f);

        // ---- windowed, mean-removed samples; zero-pad [256,288) ----
        #pragma unroll
        for (int i = 0; i < 8; ++i) {
            const int t = i * 32 + lane;
            vb[t] = (win[t] - mean) * s_hann[t];
        }
        vb[256 + lane] = 0.0f;
        __builtin_amdgcn_wave_barrier();              // LDS is in-order within a wave

        // ---- per-lane fragment registers: r[i] = v[16*(2*hi + i) + col] ----
        // A chunk q      = { r[4q],   r[4q+1] }   (A[m,k] = v[16k+m])
        // B_c chunk q    = { r[4q+c], r[4q+c+1] } (B_c[k,n] = v[16(k+c)+n])
        float r[16];
        #pragma unroll
        for (int i = 0; i < 16; ++i) r[i] = vb[(2 * hi + i) * 16 + col];

        // ---- three 16x16x16 fp32 GEMMs as chained 16x16x4 WMMAs ----
        v8f c0 = {}; v8f c1 = {}; v8f c2 = {};
        #pragma unroll
        for (int q = 0; q < 4; ++q) {
            v2f a  = { r[4 * q + 0], r[4 * q + 1] };
            v2f b1 = { r[4 * q + 1], r[4 * q + 2] };
            v2f b2 = { r[4 * q + 2], r[4 * q + 3] };
            c0 = __builtin_amdgcn_wmma_f32_16x16x4_f32(false, a, false, a,  (short)0, c0, false, false);
            c1 = __builtin_amdgcn_wmma_f32_16x16x4_f32(false, a, false, b1, (short)0, c1, false, false);
            c2 = __builtin_amdgcn_wmma_f32_16x16x4_f32(false, a, false, b2, (short)0, c2, false, false);
        }

        // ---- scatter C_c[m][n] to LDS (lane holds rows m = 8*hi + j, col n) ----
        #pragma unroll
        for (int j = 0; j < 8; ++j) {
            const int m = 8 * hi + j;
            cb[0 * CSTRIDE_C + m * CSTRIDE_M + col] = c0[j];
            cb[1 * CSTRIDE_C + m * CSTRIDE_M + col] = c1[j];
            cb[2 * CSTRIDE_C + m * CSTRIDE_M + col] = c2[j];
        }
        __builtin_amdgcn_wave_barrier();

        // ---- lane 'ofs' gathers its 16 diagonal contributions ----
        // autocorr[ofs] = sum_m C_{(ofs+m)>>4}[m, (ofs+m)&15]
        float acc = 0.0f;
        #pragma unroll
        for (int m = 0; m < 16; ++m) {
            const int s2 = lane + m;
            acc += cb[(s2 >> 4) * CSTRIDE_C + m * CSTRIDE_M + (s2 & 15)];
        }
        out[((size_t)row * NWIN + w) * NUM_AC + lane] = acc * (1.0f / 256.0f);
        __builtin_amdgcn_wave_barrier();
    }
}

extern "C" void kernel_launch(void* const* d_in, const int* in_sizes, int n_in,
                              void* d_out, int out_size, void* d_ws, size_t ws_size,
                              hipStream_t stream) {
    const float* x   = (const float*)d_in[0];
    float*       out = (float*)d_out;
    dim3 grid(16 * 128);   // one block per (batch, seq) row
    dim3 block(256);       // 8 wave32s
    hipLaunchKernelGGL(autocorr_wmma_kernel, grid, block, 0, stream, x, out);
}